// GraphConvShim_14353780704090
// MI455X (gfx1250) — compile-verified
//
#include <hip/hip_runtime.h>
#include <hip/hip_bf16.h>

// ---------------------------------------------------------------------------
// Gated GraphConv: gate -> scatter-add -> dual GEMM (WMMA f32) -> relu -> LN
// ---------------------------------------------------------------------------

#define NNODES   50000
#define NEDGES   1600000
#define IN_DIM   128
#define HID_DIM  256
#define EDGE_DIM 8
#define LN_EPS   1e-5f

typedef __attribute__((ext_vector_type(2))) float v2f;
typedef __attribute__((ext_vector_type(8))) float v8f;

// ---------------------------------------------------------------------------
// Kernel 1: zero the scatter accumulators (agg[N*128], deg[N]) every launch.
// ---------------------------------------------------------------------------
__global__ void k_zero(float* __restrict__ agg, float* __restrict__ deg,
                       int n_agg, int n_deg) {
    int i = blockIdx.x * blockDim.x + threadIdx.x;
    if (i < n_agg) agg[i] = 0.0f;
    if (i < n_deg) deg[i] = 0.0f;
}

// ---------------------------------------------------------------------------
// Kernel 2: per-edge gate w = max(sigmoid(edge_attr . W_edge), 1e-6) and
//           degree accumulation (one float atomic per edge).
// ---------------------------------------------------------------------------
__global__ void k_gate(const float* __restrict__ edge_attr,
                       const float* __restrict__ W_edge,
                       const int*   __restrict__ dst,
                       float* __restrict__ w,
                       float* __restrict__ deg,
                       int nE) {
    int e = blockIdx.x * blockDim.x + threadIdx.x;
    if (e >= nE) return;
    float s = 0.0f;
#pragma unroll
    for (int j = 0; j < EDGE_DIM; ++j)
        s += edge_attr[(size_t)e * EDGE_DIM + j] * W_edge[j];
    float g = 1.0f / (1.0f + expf(-s));
    g = fmaxf(g, 1e-6f);
    w[e] = g;
    atomicAdd(&deg[dst[e]], 1.0f);
}

// ---------------------------------------------------------------------------
// Kernel 3: gather z[src], scale by gate, scatter-add into agg[dst].
// 32 lanes per edge, float4 chunk per lane -> 4x global_atomic_add_f32.
// ---------------------------------------------------------------------------
__global__ void k_scatter(const float* __restrict__ z,
                          const int*   __restrict__ src,
                          const int*   __restrict__ dst,
                          const float* __restrict__ w,
                          float* __restrict__ agg,
                          int nE) {
    int lane = threadIdx.x & 31;
    int e = blockIdx.x * (blockDim.x >> 5) + (threadIdx.x >> 5);
    if (e >= nE) return;
    int s = src[e];
    int d = dst[e];
    float g = w[e];
    const float4 v = *(const float4*)(z + (size_t)s * IN_DIM + lane * 4);
    float* ap = agg + (size_t)d * IN_DIM + lane * 4;
    atomicAdd(ap + 0, v.x * g);
    atomicAdd(ap + 1, v.y * g);
    atomicAdd(ap + 2, v.z * g);
    atomicAdd(ap + 3, v.w * g);
}

// ---------------------------------------------------------------------------
// Kernel 4: h = z @ W_self + (agg/deg) @ W_nei ; relu ; layernorm.
// One wave -> 16 rows x 256 cols (16 x v8f accumulators).
// fp32 WMMA 16x16x4.
//
// B staging uses a K-pair interleaved LDS layout:
//   lds[m][p*PSTR + col*2 + parity]  holds  W[2p+parity][col]
// so a B fragment (k, k+1 at one col) is a single aligned ds_load_b64 into a
// contiguous even VGPR pair -> no repack movs before the WMMA.
// PSTR % 64 == 32 so the two lane halves (pair p vs p+1) hit disjoint bank
// halves: a full wave of b64 reads covers all 64 banks exactly once.
// ---------------------------------------------------------------------------
#define KCH   16
#define NPAIR (KCH / 2)              // 8 K-pairs per chunk
#define PSTR  (2 * HID_DIM + 32)     // 544 floats, %64 == 32
#define LDSOFF (NPAIR * PSTR)        // offset of W_nei plane
#define NT  (NNODES / 16)            // 3125 row tiles
#define WAVES_PER_BLK 8

__global__ void __launch_bounds__(256)
k_gemm_ln(const float* __restrict__ z,
          const float* __restrict__ agg,
          const float* __restrict__ deg,
          const float* __restrict__ W_self,
          const float* __restrict__ W_nei,
          const float* __restrict__ gamma,
          const float* __restrict__ beta,
          float* __restrict__ out) {
    __shared__ float lds[2 * NPAIR * PSTR];   // 34,816 bytes

    const int tid  = threadIdx.x;
    const int wave = tid >> 5;
    const int lane = tid & 31;
    const int lr   = lane & 15;   // col-within-tile / row-within-tile
    const int lh   = lane >> 4;   // lane half

    const int rowblk = blockIdx.x * WAVES_PER_BLK + wave;
    const bool active = (rowblk < NT);       // wave-uniform -> EXEC all-1s
    const int row0 = (active ? rowblk : 0) * 16;
    const int arow = row0 + lr;              // A-fragment source row

    float idg = 0.0f;
    if (active) idg = 1.0f / fmaxf(deg[arow], 1.0f);

    v8f acc[16];
#pragma unroll
    for (int c = 0; c < 16; ++c) acc[c] = (v8f)0.0f;

    for (int kc = 0; kc < IN_DIM / KCH; ++kc) {
        const int k0 = kc * KCH;
        __syncthreads();
        // pair-wise cooperative stage: thread -> (pair p, col c) of both W's
        for (int i = tid; i < NPAIR * HID_DIM; i += 256) {
            int p = i >> 8;           // 0..7
            int c = i & 255;          // col
            const size_t g0 = (size_t)(k0 + 2 * p) * HID_DIM + c;
            float2 s = { W_self[g0], W_self[g0 + HID_DIM] };
            float2 n = { W_nei [g0], W_nei [g0 + HID_DIM] };
            *(float2*)(&lds[p * PSTR + c * 2])          = s;
            *(float2*)(&lds[LDSOFF + p * PSTR + c * 2]) = n;
        }
        __syncthreads();

        if (active) {
#pragma unroll
            for (int kk = 0; kk < KCH; kk += 4) {
                const int p = (kk >> 1) + lh;    // this lane's K-pair
                const int ka = kk + 2 * lh;
                const float2 az = *(const float2*)(z   + (size_t)arow * IN_DIM + k0 + ka);
                const float2 an = *(const float2*)(agg + (size_t)arow * IN_DIM + k0 + ka);
                v2f a_s = { az.x, az.y };
                v2f a_n = { an.x * idg, an.y * idg };
#pragma unroll
                for (int c = 0; c < 16; ++c) {
                    const int col = c * 16 + lr;
                    const float2 bs = *(const float2*)(&lds[p * PSTR + col * 2]);
                    const float2 bn = *(const float2*)(&lds[LDSOFF + p * PSTR + col * 2]);
                    v2f b_s = { bs.x, bs.y };
                    v2f b_n = { bn.x, bn.y };
                    acc[c] = __builtin_amdgcn_wmma_f32_16x16x4_f32(
                        false, a_s, false, b_s, (short)0, acc[c], false, false);
                    acc[c] = __builtin_amdgcn_wmma_f32_16x16x4_f32(
                        false, a_n, false, b_n, (short)0, acc[c], false, false);
                }
            }
        }
    }

    if (!active) return;

    // ---- relu + per-row statistics (row v+8*lh lives at col lr per tile) ----
    float rsum[8], rsq[8];
#pragma unroll
    for (int v = 0; v < 8; ++v) { rsum[v] = 0.0f; rsq[v] = 0.0f; }
#pragma unroll
    for (int c = 0; c < 16; ++c) {
#pragma unroll
        for (int v = 0; v < 8; ++v) {
            float x = fmaxf(acc[c][v], 0.0f);
            acc[c][v] = x;
            rsum[v] += x;
            rsq[v]  += x * x;
        }
    }
    // butterfly over the 16 lanes of each half (wave32, masks 1..8)
#pragma unroll
    for (int v = 0; v < 8; ++v) {
#pragma unroll
        for (int m = 1; m < 16; m <<= 1) {
            rsum[v] += __shfl_xor(rsum[v], m, 32);
            rsq[v]  += __shfl_xor(rsq[v],  m, 32);
        }
    }
    float mean[8], rstd[8];
#pragma unroll
    for (int v = 0; v < 8; ++v) {
        mean[v] = rsum[v] * (1.0f / HID_DIM);
        float var = rsq[v] * (1.0f / HID_DIM) - mean[v] * mean[v];
        rstd[v] = rsqrtf(var + LN_EPS);
    }

    // ---- normalize + affine + store ----
#pragma unroll
    for (int c = 0; c < 16; ++c) {
        const int col = c * 16 + lr;
        const float gm = gamma[col];
        const float bt = beta[col];
#pragma unroll
        for (int v = 0; v < 8; ++v) {
            const int row = row0 + v + 8 * lh;
            out[(size_t)row * HID_DIM + col] =
                (acc[c][v] - mean[v]) * rstd[v] * gm + bt;
        }
    }
}

// ---------------------------------------------------------------------------
// Launch
// ---------------------------------------------------------------------------
extern "C" void kernel_launch(void* const* d_in, const int* in_sizes, int n_in,
                              void* d_out, int out_size, void* d_ws, size_t ws_size,
                              hipStream_t stream) {
    const float* z       = (const float*)d_in[0];
    const int*   ei      = (const int*)  d_in[1];   // [2, E] row-major
    const float* eattr   = (const float*)d_in[2];
    const float* W_self  = (const float*)d_in[3];
    const float* W_nei   = (const float*)d_in[4];
    const float* W_edge  = (const float*)d_in[5];
    const float* gamma   = (const float*)d_in[6];
    const float* beta    = (const float*)d_in[7];
    float*       out     = (float*)d_out;

    const int* src = ei;
    const int* dst = ei + NEDGES;

    float* agg = (float*)d_ws;                       // N*128
    float* deg = agg + (size_t)NNODES * IN_DIM;      // N
    float* w   = deg + NNODES;                       // E

    {
        int n_agg = NNODES * IN_DIM;
        int blocks = (n_agg + 255) / 256;
        k_zero<<<blocks, 256, 0, stream>>>(agg, deg, n_agg, NNODES);
    }
    {
        int blocks = (NEDGES + 255) / 256;
        k_gate<<<blocks, 256, 0, stream>>>(eattr, W_edge, dst, w, deg, NEDGES);
    }
    {
        int edges_per_blk = 256 / 32;
        int blocks = (NEDGES + edges_per_blk - 1) / edges_per_blk;
        k_scatter<<<blocks, 256, 0, stream>>>(z, src, dst, w, agg, NEDGES);
    }
    {
        int blocks = (NT + WAVES_PER_BLK - 1) / WAVES_PER_BLK;   // 391
        k_gemm_ln<<<blocks, 256, 0, stream>>>(z, agg, deg, W_self, W_nei,
                                              gamma, beta, out);
    }
}